// LinearGating_74629351735464
// MI455X (gfx1250) — compile-verified
//
#include <hip/hip_runtime.h>
#include <math.h>
#include <stdint.h>

typedef __attribute__((ext_vector_type(2))) float v2f;
typedef __attribute__((ext_vector_type(4))) float v4f;
typedef __attribute__((ext_vector_type(8))) float v8f;

#define N_TOK 16384
#define DDIM  2048
#define EDIM  64
#define KC    64                    // K rows per staged chunk
#define NCH   (DDIM / KC)           // 32 chunks
#define PAIRS (KC / 2)              // 32 K-row pairs per chunk
#define PSTR  80                    // padded pair-row stride (pairs of floats)
#define MATF  (PAIRS * PSTR * 2)    // floats per matrix chunk  (5120 = 20KB)
#define BUFF  (2 * MATF)            // floats per buffer (gate+noise)

// CDNA5 async global->LDS copies (ASYNCcnt)
__device__ __forceinline__ void async_g2l_b32(uint32_t lds_off, const float* g) {
  asm volatile("global_load_async_to_lds_b32 %0, %1, off"
               :: "v"(lds_off), "v"(g) : "memory");
}

// ---------------------------------------------------------------------------
// Kernel 1: dual GEMM (gate + noise) via V_WMMA_F32_16X16X4_F32.
// 8 waves/block, each wave owns TWO 16-token row tiles (block = 256 tokens).
// W chunks staged pair-interleaved into double-buffered LDS via async copies.
// ---------------------------------------------------------------------------
__global__ __launch_bounds__(256) void gating_gemm_kernel(
    const float* __restrict__ x,      // [N_TOK, DDIM]
    const float* __restrict__ Wg,     // [DDIM, EDIM]
    const float* __restrict__ Wn,     // [DDIM, EDIM]
    const float* __restrict__ noise,  // [N_TOK, EDIM]
    float* __restrict__ out_logits)   // [N_TOK, EDIM]
{
  __shared__ float smem[2 * BUFF];   // double-buffered [gate|noise], 80 KB

  const int tid    = threadIdx.x;
  const int wave   = blockIdx.x * 8 + (tid >> 5);
  const int lane   = tid & 31;
  const int lane16 = lane & 15;
  const int half   = lane >> 4;      // 0: lanes 0-15, 1: lanes 16-31
  const int tok0   = wave * 32;      // two 16-token tiles

  v8f accg[2][4] = {};
  v8f accn[2][4] = {};

  // A fragment rows (lane16 = M); this lane supplies K = k + 2*half + {0,1}
  const float* __restrict__ arow0 = x + (size_t)(tok0 + lane16) * DDIM + 2 * half;
  const float* __restrict__ arow1 = arow0 + (size_t)16 * DDIM;

  // ---- stage W chunk `ch` into buffer `b`, pair-interleaved ----------------
  // element (r,c) of the chunk -> LDS float index ((r>>1)*PSTR + c)*2 + (r&1)
  auto stage = [&](int ch, int b) {
    const int kbase = ch * KC;
    const float* __restrict__ gG = Wg + (size_t)kbase * EDIM;
    const float* __restrict__ gN = Wn + (size_t)kbase * EDIM;
    float* const baseG = &smem[b * BUFF];
    float* const baseN = baseG + MATF;
#pragma unroll
    for (int i = 0; i < 16; ++i) {
      const int idx = i * 256 + tid;           // 0..4095
      const int r = idx >> 6;                  // chunk row 0..63
      const int c = idx & 63;                  // col 0..63
      const int fi = ((r >> 1) * PSTR + c) * 2 + (r & 1);
      async_g2l_b32((uint32_t)(uintptr_t)&baseG[fi], gG + idx);
      async_g2l_b32((uint32_t)(uintptr_t)&baseN[fi], gN + idx);
    }
  };

  stage(0, 0);
  int buf = 0;
  for (int ch = 0; ch < NCH; ++ch) {
    if (ch + 1 < NCH) {
      stage(ch + 1, buf ^ 1);                      // prefetch next chunk
      asm volatile("s_wait_asynccnt 0x20" ::: "memory");  // <=32: cur chunk done
    } else {
      asm volatile("s_wait_asynccnt 0x0" ::: "memory");
    }
    __syncthreads();

    const float* __restrict__ bufG = &smem[buf * BUFF];
    const float* __restrict__ bufN = bufG + MATF;
    const int kbase = ch * KC;

    for (int kk = 0; kk < KC; kk += 4) {
      const v2f a0 = *(const v2f*)(arow0 + kbase + kk);
      const v2f a1 = *(const v2f*)(arow1 + kbase + kk);
      const int p = (kk >> 1) + half;              // K-row pair for this lane
#pragma unroll
      for (int g = 0; g < 4; ++g) {
        const int pe = (p * PSTR + g * 16 + lane16) * 2;
        const v2f bg = *(const v2f*)&bufG[pe];     // single ds_load_b64
        const v2f bn = *(const v2f*)&bufN[pe];
        accg[0][g] = __builtin_amdgcn_wmma_f32_16x16x4_f32(
            false, a0, false, bg, (short)0, accg[0][g], false, false);
        accg[1][g] = __builtin_amdgcn_wmma_f32_16x16x4_f32(
            false, a1, false, bg, (short)0, accg[1][g], false, false);
        accn[0][g] = __builtin_amdgcn_wmma_f32_16x16x4_f32(
            false, a0, false, bn, (short)0, accn[0][g], false, false);
        accn[1][g] = __builtin_amdgcn_wmma_f32_16x16x4_f32(
            false, a1, false, bn, (short)0, accn[1][g], false, false);
      }
    }
    __syncthreads();                               // buffer read complete (WAR)
    buf ^= 1;
  }

  // ---- epilogue: logits = gate + noise * softplus(noise_logit)/(E*E) ------
  // C layout: VGPR r of lane -> row M = r + 8*half, col N = lane16 (+16*g)
#pragma unroll
  for (int tile = 0; tile < 2; ++tile) {
#pragma unroll
    for (int g = 0; g < 4; ++g) {
#pragma unroll
      for (int r = 0; r < 8; ++r) {
        const int t  = tok0 + tile * 16 + r + 8 * half;
        const int ec = g * 16 + lane16;
        const float nl = accn[tile][g][r];
        const float sp = fmaxf(nl, 0.0f) + log1pf(expf(-fabsf(nl)));
        const float lg = accg[tile][g][r] +
                         noise[(size_t)t * EDIM + ec] * (sp * (1.0f / 4096.0f));
        out_logits[(size_t)t * EDIM + ec] = lg;
      }
    }
  }
}

// ---------------------------------------------------------------------------
// Kernel 2: per-token top-2 + masked softmax + raw softmax.
// ---------------------------------------------------------------------------
__global__ __launch_bounds__(256) void gating_topk_kernel(
    const float* __restrict__ logits,   // [N_TOK, EDIM]
    float* __restrict__ weights,        // [N_TOK, EDIM]
    int* __restrict__ indices,          // [N_TOK, 2]
    float* __restrict__ raw_probs)      // [N_TOK, EDIM]
{
  const int t = blockIdx.x * blockDim.x + threadIdx.x;
  if (t >= N_TOK) return;
  const float* __restrict__ row = logits + (size_t)t * EDIM;

  float v[EDIM];
#pragma unroll
  for (int i = 0; i < EDIM / 4; ++i) {
    const v4f q = *(const v4f*)(row + 4 * i);
    v[4 * i + 0] = q.x; v[4 * i + 1] = q.y;
    v[4 * i + 2] = q.z; v[4 * i + 3] = q.w;
  }

  float m = v[0];
#pragma unroll
  for (int e = 1; e < EDIM; ++e) m = fmaxf(m, v[e]);

  // top-2 with jax.lax.top_k tie-breaking (earliest index wins via strict >)
  float v1 = -INFINITY, v2 = -INFINITY;
  int i1 = 0, i2 = 0;
#pragma unroll
  for (int e = 0; e < EDIM; ++e) {
    const float val = v[e];
    if (val > v1)      { v2 = v1; i2 = i1; v1 = val; i1 = e; }
    else if (val > v2) { v2 = val; i2 = e; }
  }

  float sum = 0.0f;
#pragma unroll
  for (int e = 0; e < EDIM; ++e) sum += expf(v[e] - m);
  const float inv = 1.0f / sum;

  // masked softmax: non-top entries are exactly 0 (exp(-1e9 - max) -> 0 in f32)
  const float e2 = expf(v2 - v1);
  const float w1 = 1.0f / (1.0f + e2);
  const float w2 = e2 * w1;

#pragma unroll
  for (int e = 0; e < EDIM; ++e) {
    const float w = (e == i1) ? w1 : ((e == i2) ? w2 : 0.0f);
    weights[(size_t)t * EDIM + e]   = w;
    raw_probs[(size_t)t * EDIM + e] = expf(v[e] - m) * inv;
  }
  indices[(size_t)t * 2 + 0] = i1;
  indices[(size_t)t * 2 + 1] = i2;
}

// ---------------------------------------------------------------------------
extern "C" void kernel_launch(void* const* d_in, const int* in_sizes, int n_in,
                              void* d_out, int out_size, void* d_ws,
                              size_t ws_size, hipStream_t stream) {
  const float* x     = (const float*)d_in[0];  // [4,4096,2048]
  const float* Wg    = (const float*)d_in[1];  // [2048,64]
  const float* Wn    = (const float*)d_in[2];  // [2048,64]
  const float* noise = (const float*)d_in[3];  // [16384,64]
  // d_in[4] = top_k scalar on device; fixed to 2 (cannot sync-read in capture)

  float* out      = (float*)d_out;
  float* weights  = out;                                             // N*64
  int*   indices  = (int*)(out + (size_t)N_TOK * EDIM);              // N*2 int32
  float* logitsO  = out + (size_t)N_TOK * EDIM + (size_t)N_TOK * 2;  // N*64
  float* rawp     = logitsO + (size_t)N_TOK * EDIM;                  // N*64

  // Kernel 1: 512 waves, one per 32 tokens; 8 waves per 256-thread block
  dim3 blkA(256), grdA((N_TOK / 32) / 8);  // 64 blocks
  gating_gemm_kernel<<<grdA, blkA, 0, stream>>>(x, Wg, Wn, noise, logitsO);

  // Kernel 2: one thread per token
  dim3 blkB(256), grdB((N_TOK + 255) / 256);
  gating_topk_kernel<<<grdB, blkB, 0, stream>>>(logitsO, weights, indices, rawp);
}